// RegressionGCN_44049184588188
// MI455X (gfx1250) — compile-verified
//
#include <hip/hip_runtime.h>
#include <stdint.h>

// CDNA5 fp32 WMMA operand types: A/B are 16x4 / 4x16 fp32 -> 2 VGPRs/lane,
// C/D is 16x16 fp32 -> 8 VGPRs/lane (wave32).
typedef __attribute__((ext_vector_type(2))) float v2f;
typedef __attribute__((ext_vector_type(8))) float v8f;

// ---------------------------------------------------------------------------
// Degree / dinv
// ---------------------------------------------------------------------------
__global__ void k_fill1(float* __restrict__ p, int n) {
  int i = blockIdx.x * blockDim.x + threadIdx.x;
  if (i < n) p[i] = 1.0f;  // self-loop contributes 1 to every degree
}

__global__ void k_deg(const int* __restrict__ dst, int E, float* __restrict__ deg) {
  int e = blockIdx.x * blockDim.x + threadIdx.x;
  if (e < E) atomicAdd(&deg[dst[e]], 1.0f);
}

__global__ void k_rsqrt_inplace(float* __restrict__ p, int n) {
  int i = blockIdx.x * blockDim.x + threadIdx.x;
  if (i < n) p[i] = rsqrtf(p[i]);  // deg >= 1 always
}

// ---------------------------------------------------------------------------
// H = X @ W via V_WMMA_F32_16X16X4_F32 (exact fp32 matrix pipe).
// One wave per 16-row M tile; NT 16-col N tiles per wave (A reused).
// A-layout: lane L holds row M=L%16; VGPR pair = K = k0+2*(L>=16), +1.
// B staged in LDS transposed (Wt[col][k], stride KS=KPAD+4: bank-conflict-free,
// zero-padded for col>=F / k>=KREAL) -> inner loop is branch-free:
//   global_load_b64 (A) + ds_load_b64 (B) + v_wmma per tile.
// K-padding of A beyond KREAL relies on the caller guaranteeing X's padded
// feature columns are zero (the init kernels below maintain this invariant).
// Strides LDX/LDH are compile-time -> stores use one base addr + imm offsets.
// ---------------------------------------------------------------------------
template <int KREAL, int KPAD, int F, int NT, int LDX, int LDH>
__global__ void k_gemm_wmma_f32(const float* __restrict__ X,
                                const float* __restrict__ W,
                                float* __restrict__ H,
                                int nNodes) {
  constexpr int NCOLS = NT * 16;
  constexpr int KS = KPAD + 4;  // LDS row stride (even, +4 -> distinct bank pairs)
  __shared__ float Wt[NCOLS * KS];

  // Cooperative staging of W (transposed, zero-padded). Runs once per block.
  for (int idx = threadIdx.x; idx < NCOLS * KS; idx += blockDim.x) {
    int c = idx / KS;
    int k = idx - c * KS;
    Wt[idx] = (c < F && k < KREAL) ? W[k * F + c] : 0.0f;
  }
  __syncthreads();

  int wave = blockIdx.x * (blockDim.x >> 5) + (threadIdx.x >> 5);
  int mTiles = (nNodes + 15) >> 4;
  if (wave >= mTiles) return;  // wave-uniform: EXEC all-1 at WMMA

  int lane = threadIdx.x & 31;
  int half = lane >> 4;   // 0: K pair k0..k0+1, 1: K pair k0+2..k0+3
  int l15  = lane & 15;
  int row = (wave << 4) + l15;
  if (row >= nNodes) row = nNodes - 1;  // clamp loads; stores are guarded
  const float* xrow = X + (size_t)row * LDX;

  v8f acc[NT];
#pragma unroll
  for (int t = 0; t < NT; ++t) acc[t] = (v8f){};

#pragma unroll 4
  for (int k0 = 0; k0 < KPAD; k0 += 4) {
    int ka = k0 + (half << 1);
    v2f a = *(const v2f*)(xrow + ka);  // 8B-aligned: ka even, LDX even
#pragma unroll
    for (int t = 0; t < NT; ++t) {
      v2f b = *(const v2f*)(&Wt[(t * 16 + l15) * KS + ka]);
      // (neg_a, A, neg_b, B, c_mod, C, reuse_a, reuse_b)
      acc[t] = __builtin_amdgcn_wmma_f32_16x16x4_f32(false, a, false, b, (short)0,
                                                     acc[t], false, false);
    }
  }

  // Store epilogue. C/D layout: VGPR r -> (M = r + 8*half, N = l15) per tile.
  int mbase = (wave << 4) + (half << 3);
  bool fullTile = ((wave << 4) + 16) <= nNodes;  // wave-uniform
#pragma unroll
  for (int t = 0; t < NT; ++t) {
    float* hp = H + (size_t)mbase * LDH + t * 16 + l15;
    if (fullTile) {
#pragma unroll
      for (int r = 0; r < 8; ++r) hp[r * LDH] = acc[t][r];  // imm offsets, coalesced
    } else {
#pragma unroll
      for (int r = 0; r < 8; ++r)
        if (mbase + r < nNodes) hp[r * LDH] = acc[t][r];
    }
  }
}

// ---------------------------------------------------------------------------
// O[i,f] = self-loop term = H[i,f] * dinv[i]^2 for f<F; 0 for F<=f<FPAD.
// (The zero padding is what lets the next layer's GEMM read A unguarded.)
// ---------------------------------------------------------------------------
template <int F, int FPAD>
__global__ void k_selfloop_init(const float* __restrict__ H, int ldh,
                                const float* __restrict__ dinv,
                                float* __restrict__ O, int ldo, int nNodes) {
  int t = blockIdx.x * blockDim.x + threadIdx.x;
  int i = t / FPAD;          // FPAD is a power of two -> shift
  int f = t - i * FPAD;
  if (i >= nNodes) return;
  float v = 0.0f;
  if (f < F) {
    float d = dinv[i];
    v = H[(size_t)i * ldh + f] * d * d;
  }
  O[(size_t)i * ldo + f] = v;
}

// ---------------------------------------------------------------------------
// Edge aggregation: O[dst,f] += H[src,f] * dinv[src]*dinv[dst].
// FPAD lanes per edge -> coalesced gather + coalesced float atomics.
// ---------------------------------------------------------------------------
template <int F, int FPAD>
__global__ void k_edge_scatter(const int* __restrict__ src, const int* __restrict__ dst,
                               int E,
                               const float* __restrict__ H, int ldh,
                               const float* __restrict__ dinv,
                               float* __restrict__ O, int ldo) {
  int t = blockIdx.x * blockDim.x + threadIdx.x;
  int e = t / FPAD;
  int f = t - e * FPAD;
  if (e >= E || f >= F) return;
  int s = src[e];
  int d = dst[e];
  float nrm = dinv[s] * dinv[d];
  atomicAdd(&O[(size_t)d * ldo + f], H[(size_t)s * ldh + f] * nrm);
}

// ---------------------------------------------------------------------------
// O[i,f] = act(O[i,f] + b[f]) for f < F only (padding stays 0).
// ---------------------------------------------------------------------------
template <int F, bool RELU>
__global__ void k_bias_act(float* __restrict__ O, int ldo,
                           const float* __restrict__ b, int nNodes) {
  int t = blockIdx.x * blockDim.x + threadIdx.x;
  int i = t / F;
  int f = t - i * F;
  if (i >= nNodes) return;
  float v = O[(size_t)i * ldo + f] + b[f];
  if (RELU) v = fmaxf(v, 0.0f);
  O[(size_t)i * ldo + f] = v;
}

// ---------------------------------------------------------------------------
// Orchestration
// ---------------------------------------------------------------------------
extern "C" void kernel_launch(void* const* d_in, const int* in_sizes, int n_in,
                              void* d_out, int out_size, void* d_ws, size_t ws_size,
                              hipStream_t stream) {
  const float* x  = (const float*)d_in[0];
  const int*   ei = (const int*)d_in[1];   // edge_index [2,E], int32 per harness convention
  const float* W1 = (const float*)d_in[2];
  const float* b1 = (const float*)d_in[3];
  const float* W2 = (const float*)d_in[4];
  const float* b2 = (const float*)d_in[5];
  const float* W3 = (const float*)d_in[6];
  const float* b3 = (const float*)d_in[7];
  float* out = (float*)d_out;

  const int nNodes = in_sizes[0] / 128;
  const int E = in_sizes[1] / 2;
  const int* src = ei;
  const int* dst = ei + E;

  // Workspace layout: dinv | hA (N x 32) | hB (N x 32)  (~26 MB total)
  char* ws = (char*)d_ws;
  float* dinv = (float*)ws;
  size_t offA = ((size_t)nNodes * sizeof(float) + 255) & ~(size_t)255;
  float* hA = (float*)(ws + offA);
  size_t offB = offA + (((size_t)nNodes * 32 * sizeof(float) + 255) & ~(size_t)255);
  float* hB = (float*)(ws + offB);

  const int B = 256;
  const int mTiles = (nNodes + 15) / 16;
  auto cdiv = [](long long a, long long b) { return (unsigned)((a + b - 1) / b); };

  // dinv = rsqrt(1 + in_degree)
  k_fill1<<<cdiv(nNodes, B), B, 0, stream>>>(dinv, nNodes);
  k_deg<<<cdiv(E, B), B, 0, stream>>>(dst, E, dinv);
  k_rsqrt_inplace<<<cdiv(nNodes, B), B, 0, stream>>>(dinv, nNodes);

  // ---- Layer 1: K=128, F=25 (NT=2 -> cols 0..31, cols>=25 exactly 0), relu ----
  k_gemm_wmma_f32<128, 128, 25, 2, 128, 32><<<cdiv(mTiles, 4), 128, 0, stream>>>(
      x, W1, hA, nNodes);
  k_selfloop_init<25, 32><<<cdiv((long long)nNodes * 32, B), B, 0, stream>>>(
      hA, 32, dinv, hB, 32, nNodes);
  k_edge_scatter<25, 32><<<cdiv((long long)E * 32, B), B, 0, stream>>>(
      src, dst, E, hA, 32, dinv, hB, 32);
  k_bias_act<25, true><<<cdiv((long long)nNodes * 25, B), B, 0, stream>>>(
      hB, 32, b1, nNodes);

  // ---- Layer 2: K=25 (pad 28; hB cols 25..27 are 0), F=10, relu ----
  k_gemm_wmma_f32<25, 28, 10, 1, 32, 32><<<cdiv(mTiles, 4), 128, 0, stream>>>(
      hB, W2, hA, nNodes);
  k_selfloop_init<10, 16><<<cdiv((long long)nNodes * 16, B), B, 0, stream>>>(
      hA, 32, dinv, hB, 32, nNodes);
  k_edge_scatter<10, 16><<<cdiv((long long)E * 16, B), B, 0, stream>>>(
      src, dst, E, hA, 32, dinv, hB, 32);
  k_bias_act<10, true><<<cdiv((long long)nNodes * 10, B), B, 0, stream>>>(
      hB, 32, b2, nNodes);

  // ---- Layer 3: K=10 (pad 12; hB cols 10..11 are 0), F=8, no relu, to d_out ----
  k_gemm_wmma_f32<10, 12, 8, 1, 32, 32><<<cdiv(mTiles, 4), 128, 0, stream>>>(
      hB, W3, hA, nNodes);
  k_selfloop_init<8, 8><<<cdiv((long long)nNodes * 8, B), B, 0, stream>>>(
      hA, 32, dinv, out, 8, nNodes);
  k_edge_scatter<8, 8><<<cdiv((long long)E * 8, B), B, 0, stream>>>(
      src, dst, E, hA, 32, dinv, out, 8);
  k_bias_act<8, false><<<cdiv((long long)nNodes * 8, B), B, 0, stream>>>(
      out, 8, b3, nNodes);
}